// TopKRouter_39436389712278
// MI455X (gfx1250) — compile-verified
//
#include <hip/hip_runtime.h>
#include <stdint.h>

#define T_TOK   16384
#define D_DIM   2048
#define E_EXP   64
#define TOPK    8

#define ROWS_PER_BLOCK 64
#define THREADS        128          // 4 wave32s
#define KB             64           // K tile
#define NTILES         (D_DIM / KB) // 32
#define STRIDE         68           // padded row stride in floats (272B, 16B aligned)

#define XS_FLOATS (ROWS_PER_BLOCK * STRIDE)   // 4352 floats = 17408 B
#define WS_FLOATS (E_EXP * STRIDE)            // 4352 floats = 17408 B
#define XS_BYTES  (XS_FLOATS * 4)
#define WS_BYTES  (WS_FLOATS * 4)
#define BUF_BYTES (XS_BYTES + WS_BYTES)       // 34816 B per buffer
// layout: buf0.xs | buf0.ws | buf1.xs | buf1.ws | cnt[64] | bias[64]
#define CNT_OFF   (2 * BUF_BYTES)
#define SMEM_BYTES (2 * BUF_BYTES + E_EXP * 4 + E_EXP * 4)

typedef float v2f __attribute__((ext_vector_type(2)));
typedef float v8f __attribute__((ext_vector_type(8)));

// CDNA5 async load: memory -> LDS, tracked by ASYNCcnt (in-order completion)
__device__ __forceinline__ void async_copy_b128(uint32_t lds_off, const void* gptr) {
    uint64_t ga = (uint64_t)gptr;
    asm volatile("global_load_async_to_lds_b128 %0, %1, off"
                 :: "v"(lds_off), "v"(ga) : "memory");
}
// Non-temporal flavor for the streamed-once x tensor (keep gate_w hot in L2)
__device__ __forceinline__ void async_copy_b128_nt(uint32_t lds_off, const void* gptr) {
    uint64_t ga = (uint64_t)gptr;
    asm volatile("global_load_async_to_lds_b128 %0, %1, off th:TH_LOAD_NT"
                 :: "v"(lds_off), "v"(ga) : "memory");
}

// Issue one K-tile (x: 64 rows x 64 f32, w: 64 experts x 64 f32) into buffer at buf_base.
// 16 async instructions per wave -> s_wait_asynccnt 16 after issuing the NEXT tile
// guarantees THIS tile has landed.
__device__ __forceinline__ void stage_tile(const float* __restrict__ x,
                                           const float* __restrict__ gw,
                                           int row0, int k0, uint32_t buf_base, int tid) {
#pragma unroll
    for (int i = tid; i < ROWS_PER_BLOCK * (KB / 4); i += THREADS) {   // 8 per thread
        int r = i >> 4, c = i & 15;
        async_copy_b128_nt(buf_base + (uint32_t)(r * STRIDE + c * 4) * 4u,
                           x + (size_t)(row0 + r) * D_DIM + k0 + c * 4);
    }
#pragma unroll
    for (int i = tid; i < E_EXP * (KB / 4); i += THREADS) {            // 8 per thread
        int e = i >> 4, c = i & 15;
        async_copy_b128(buf_base + (uint32_t)XS_BYTES + (uint32_t)(e * STRIDE + c * 4) * 4u,
                        gw + (size_t)e * D_DIM + k0 + c * 4);
    }
}

__global__ void __launch_bounds__(THREADS)
router_topk_kernel(const float* __restrict__ x,      // [T, D]
                   const float* __restrict__ gw,     // [E, D]
                   const float* __restrict__ bias,   // [E]
                   int*   __restrict__ oidx,         // [T, 8]
                   float* __restrict__ ow,           // [T, 8]
                   float* __restrict__ ocnt) {       // [E]
    extern __shared__ char smem[];                   // dynamic LDS, offset 0
    int*   cnt = (int*)(smem + CNT_OFF);
    float* bsh = (float*)(cnt + E_EXP);

    const int tid  = threadIdx.x;
    const int wave = tid >> 5;
    const int lane = tid & 31;
    const int hi   = lane >> 4;                      // half-wave select (K offset)
    const int n    = lane & 15;                      // M for A-frag, N for B/C-frag
    const int row0 = blockIdx.x * ROWS_PER_BLOCK;

    if (tid < E_EXP) { cnt[tid] = 0; bsh[tid] = bias[tid]; }

    const v8f vzero = {0.f,0.f,0.f,0.f,0.f,0.f,0.f,0.f};
    v8f acc[4];
#pragma unroll
    for (int t = 0; t < 4; ++t) acc[t] = vzero;

    // software pipeline prologue: tile 0 -> buffer 0
    stage_tile(x, gw, row0, 0, 0u, tid);

    for (int it = 0; it < NTILES; ++it) {
        const uint32_t cur = (uint32_t)(it & 1) * BUF_BYTES;
        if (it + 1 < NTILES) {
            // issue next tile into the other buffer, then wait only for THIS tile
            stage_tile(x, gw, row0, (it + 1) * KB,
                       (uint32_t)((it + 1) & 1) * BUF_BYTES, tid);
            asm volatile("s_wait_asynccnt 16" ::: "memory");
        } else {
            asm volatile("s_wait_asynccnt 0" ::: "memory");
        }
        __syncthreads();                             // all waves' tile-it loads visible

        const float* xs = (const float*)(smem + cur);
        const float* ws = (const float*)(smem + cur + XS_BYTES);

        // WMMA f32 16x16x4: A rows = this wave's 16 tokens, B = 4 tiles of 16 experts
        const float* xrow = xs + (wave * 16 + n) * STRIDE;
#pragma unroll 4
        for (int kk = 0; kk < KB; kk += 4) {
            v2f a = *(const v2f*)(xrow + kk + 2 * hi);           // K = kk + v + 2*hi, M = n
#pragma unroll
            for (int t = 0; t < 4; ++t) {
                v2f b = *(const v2f*)(ws + (t * 16 + n) * STRIDE + kk + 2 * hi);
                acc[t] = __builtin_amdgcn_wmma_f32_16x16x4_f32(
                    false, a, false, b, (short)0, acc[t], false, false);
            }
        }
        __syncthreads();   // readers done before buffer `cur` is re-written at it+1
    }

    // Spill logits to LDS (buf0.xs region; 16-row slabs are wave-private).
    // C layout: VGPR r, lanes 0-15 -> (M=r, N=lane); lanes 16-31 -> (M=r+8, N=lane-16)
    float* lg = (float*)smem;
#pragma unroll
    for (int t = 0; t < 4; ++t)
#pragma unroll
        for (int r = 0; r < 8; ++r)
            lg[(wave * 16 + r + 8 * hi) * STRIDE + t * 16 + n] = acc[t][r];

    if (lane < 16) {                                 // one lane per token row
        const int row = wave * 16 + n;
        float v[E_EXP];
#pragma unroll
        for (int j = 0; j < E_EXP; ++j) v[j] = lg[row * STRIDE + j] + bsh[j];

        int      bidx[TOPK];
        float    bval[TOPK];
        uint64_t chosen = 0;
#pragma unroll
        for (int kth = 0; kth < TOPK; ++kth) {
            float bv = -3.402823466e38f;
            int   bi = 0;
#pragma unroll
            for (int j = 0; j < E_EXP; ++j) {
                bool ok = (((chosen >> j) & 1ull) == 0) && (v[j] > bv); // strict > : lowest index wins ties
                bv = ok ? v[j] : bv;
                bi = ok ? j    : bi;
            }
            chosen |= (1ull << bi);
            bidx[kth] = bi;
            bval[kth] = bv;
        }
        float m = bval[0], s = 0.f, w8[TOPK];
#pragma unroll
        for (int kth = 0; kth < TOPK; ++kth) { w8[kth] = __expf(bval[kth] - m); s += w8[kth]; }
        float inv = 1.0f / s;

        const int grow = row0 + row;
#pragma unroll
        for (int kth = 0; kth < TOPK; ++kth) {
            oidx[grow * TOPK + kth] = bidx[kth];
            ow[grow * TOPK + kth]   = w8[kth] * inv;
            atomicAdd(&cnt[bidx[kth]], 1);
        }
    }
    __syncthreads();
    if (tid < E_EXP) atomicAdd(&ocnt[tid], (float)cnt[tid]);   // global_atomic_add_f32
}

__global__ void init_counts_kernel(float* __restrict__ ocnt) {
    if (threadIdx.x < E_EXP) ocnt[threadIdx.x] = 0.0f;
}

extern "C" void kernel_launch(void* const* d_in, const int* in_sizes, int n_in,
                              void* d_out, int out_size, void* d_ws, size_t ws_size,
                              hipStream_t stream) {
    (void)in_sizes; (void)n_in; (void)d_ws; (void)ws_size; (void)out_size;
    const float* x    = (const float*)d_in[0];   // [4,4096,2048] f32
    const float* gw   = (const float*)d_in[1];   // [64,2048]     f32
    const float* bias = (const float*)d_in[2];   // [64]          f32
    // d_in[3] = top_k (always 8; hardcoded)

    int*   oidx = (int*)d_out;                              // [T,8] int32
    float* ow   = (float*)d_out + (size_t)T_TOK * TOPK;     // [T,8] f32
    float* ocnt = (float*)d_out + 2 * (size_t)T_TOK * TOPK; // [64]  f32

    init_counts_kernel<<<1, 64, 0, stream>>>(ocnt);
    router_topk_kernel<<<T_TOK / ROWS_PER_BLOCK, THREADS, SMEM_BYTES, stream>>>(
        x, gw, bias, oidx, ow, ocnt);
}